// GNN_Embed_69733089018032
// MI455X (gfx1250) — compile-verified
//
#include <hip/hip_runtime.h>
#include <cstddef>

#define DIN 128
#define HID 64
#define HEADS 4
#define HC 256          // HEADS*HID
#define EMB 128
#define NEG_SLOPE 0.2f
#define BN_EPS 1e-5f
#define ATT_ROW 129     // 2*HID+1
#define ATT_L (HEADS*ATT_ROW)

typedef __attribute__((ext_vector_type(2))) float v2f;
typedef __attribute__((ext_vector_type(8))) float v8f;

// ---------- order-preserving float <-> uint map for atomic segment-max ----------
__device__ __forceinline__ unsigned fmapu(float x) {
  unsigned u = __float_as_uint(x);
  return (u & 0x80000000u) ? ~u : (u | 0x80000000u);
}
__device__ __forceinline__ float funmap(unsigned u) {
  return __uint_as_float((u & 0x80000000u) ? (u & 0x7FFFFFFFu) : ~u);
}

// ---------- generic fill ----------
__global__ void fill_u32(unsigned* __restrict__ p, unsigned v, size_t n) {
  size_t i = (size_t)blockIdx.x * blockDim.x + threadIdx.x;
  size_t stride = (size_t)gridDim.x * blockDim.x;
  for (; i < n; i += stride) p[i] = v;
}

// ---------- fp32 WMMA GEMM ----------
// C[M,Nc] = (BN ? A*scale+shift : A)[M,K] @ B[K,Nc] (+bias)
// Each wave computes a 16x64 strip: 4 accumulators share one A fragment per
// K-step -> 4 v_wmma_f32_16x16x4_f32 per loop iteration, branch-free body.
template <bool BN>
__global__ __launch_bounds__(256) void gemm_wmma_f32(
    const float* __restrict__ A, const float* __restrict__ B,
    float* __restrict__ C, int M, int K, int Nc,
    const float* __restrict__ scale, const float* __restrict__ shift,
    const float* __restrict__ bias)
{
  const int wave = threadIdx.x >> 5;
  const int lane = threadIdx.x & 31;
  const int strips = Nc >> 6;               // 64-wide strips (4 for 256, 2 for 128)
  const int rowsPerBlk = 8 / strips;        // row-tiles per block
  const int st = wave % strips;
  const int mt = blockIdx.x * rowsPerBlk + wave / strips;
  const int m0 = mt * 16;
  if (m0 >= M) return;
  const int n0 = st * 64;
  const int ml = lane & 15;                 // M (for A) / N (for B,C) within tile
  const int kh = (lane >> 4) << 1;          // K sub-offset: 0 or 2
  const int mrow = (m0 + ml < M) ? (m0 + ml) : (M - 1);
  const float* Arow = A + (size_t)mrow * K;

  v8f acc0 = {}, acc1 = {}, acc2 = {}, acc3 = {};
  for (int k0 = 0; k0 < K; k0 += 4) {
    const int ka = k0 + kh;
    float a0 = Arow[ka];
    float a1 = Arow[ka + 1];
    if (BN) {
      a0 = fmaf(a0, scale[ka],     shift[ka]);
      a1 = fmaf(a1, scale[ka + 1], shift[ka + 1]);
    }
    v2f a; a[0] = a0; a[1] = a1;
    const float* B0 = B + (size_t)ka * Nc + n0 + ml;
    const float* B1 = B + (size_t)(ka + 1) * Nc + n0 + ml;
    v2f b0, b1, b2, b3;
    b0[0] = B0[0];  b0[1] = B1[0];
    b1[0] = B0[16]; b1[1] = B1[16];
    b2[0] = B0[32]; b2[1] = B1[32];
    b3[0] = B0[48]; b3[1] = B1[48];
    acc0 = __builtin_amdgcn_wmma_f32_16x16x4_f32(false, a, false, b0, (short)0, acc0, false, false);
    acc1 = __builtin_amdgcn_wmma_f32_16x16x4_f32(false, a, false, b1, (short)0, acc1, false, false);
    acc2 = __builtin_amdgcn_wmma_f32_16x16x4_f32(false, a, false, b2, (short)0, acc2, false, false);
    acc3 = __builtin_amdgcn_wmma_f32_16x16x4_f32(false, a, false, b3, (short)0, acc3, false, false);
  }

  const int mh = (lane >> 4) << 3;          // 0 or 8 (C row offset for upper lanes)
  #pragma unroll
  for (int t = 0; t < 4; ++t) {
    const v8f accv = (t == 0) ? acc0 : (t == 1) ? acc1 : (t == 2) ? acc2 : acc3;
    const int nc = n0 + t * 16 + ml;
    const float bb = bias ? bias[nc] : 0.f;
    #pragma unroll
    for (int v = 0; v < 8; ++v) {
      const int mr = m0 + mh + v;
      if (mr < M) C[(size_t)mr * Nc + nc] = accv[v] + bb;
    }
  }
}

// ---------- BatchNorm: per-feature sum / sumsq over nodes ----------
__global__ void bn_reduce(const float* __restrict__ x, float* __restrict__ sums, int N) {
  const int f = threadIdx.x;           // HID threads
  float s = 0.f, s2 = 0.f;
  for (int n = blockIdx.x; n < N; n += gridDim.x) {
    float v = x[(size_t)n * HID + f];
    s += v;
    s2 = fmaf(v, v, s2);
  }
  atomicAdd(&sums[f], s);
  atomicAdd(&sums[HID + f], s2);
}

__global__ void bn_final(const float* __restrict__ sums,
                         const float* __restrict__ gamma, const float* __restrict__ beta,
                         float* __restrict__ scale, float* __restrict__ shift, int N) {
  const int f = threadIdx.x;
  if (f >= HID) return;
  const float inv_n = 1.f / (float)N;
  const float mean = sums[f] * inv_n;
  const float var = sums[HID + f] * inv_n - mean * mean;
  const float sc = gamma[f] * rsqrtf(var + BN_EPS);
  scale[f] = sc;
  shift[f] = beta[f] - mean * sc;
}

// ---------- per-node attention scores s_i, s_j ----------
__global__ void node_s(const float* __restrict__ h, const float* __restrict__ att,
                       float* __restrict__ s_i, float* __restrict__ s_j, int N) {
  const int n = blockIdx.x * blockDim.x + threadIdx.x;
  if (n >= N) return;
  const float* hr = h + (size_t)n * HC;
  #pragma unroll
  for (int hh = 0; hh < HEADS; ++hh) {
    const float* ai = att + hh * ATT_ROW;        // att_i (uniform -> scalar loads)
    const float* aj = ai + HID;                  // att_j
    const float4* hv4 = (const float4*)(hr + hh * HID);
    float si = 0.f, sj = 0.f;
    #pragma unroll 4
    for (int c4 = 0; c4 < HID / 4; ++c4) {
      const float4 v = hv4[c4];
      const int c = c4 * 4;
      si = fmaf(v.x, ai[c],     si); sj = fmaf(v.x, aj[c],     sj);
      si = fmaf(v.y, ai[c + 1], si); sj = fmaf(v.y, aj[c + 1], sj);
      si = fmaf(v.z, ai[c + 2], si); sj = fmaf(v.z, aj[c + 2], sj);
      si = fmaf(v.w, ai[c + 3], si); sj = fmaf(v.w, aj[c + 3], sj);
    }
    s_i[(size_t)n * HEADS + hh] = si;
    s_j[(size_t)n * HEADS + hh] = sj;
  }
}

// ---------- edge pass 1: segment-max of leaky-relu logits over col ----------
__global__ void edge_max(const int* __restrict__ rows, const int* __restrict__ cols,
                         const float* __restrict__ ea,
                         const float* __restrict__ s_i, const float* __restrict__ s_j,
                         const float* __restrict__ linE, const float* __restrict__ att,
                         unsigned* __restrict__ mseg, int E) {
  const int e = blockIdx.x * blockDim.x + threadIdx.x;
  if (e >= E) return;
  const int r = rows[e], c = cols[e];
  const float w = ea[e];
  float si4[4], sj4[4];
  *(float4*)si4 = ((const float4*)s_i)[r];
  *(float4*)sj4 = ((const float4*)s_j)[c];
  #pragma unroll
  for (int hh = 0; hh < HEADS; ++hh) {
    const float coef = linE[hh] * att[hh * ATT_ROW + 2 * HID];  // uniform
    float lg = si4[hh] + sj4[hh] + w * coef;
    lg = (lg >= 0.f) ? lg : NEG_SLOPE * lg;
    atomicMax(&mseg[(size_t)c * HEADS + hh], fmapu(lg));
  }
}

// ---------- edge pass 2: p = exp(logit - max), store p, segment-sum denom ----------
__global__ void edge_p(const int* __restrict__ rows, const int* __restrict__ cols,
                       const float* __restrict__ ea,
                       const float* __restrict__ s_i, const float* __restrict__ s_j,
                       const float* __restrict__ linE, const float* __restrict__ att,
                       const unsigned* __restrict__ mseg,
                       float* __restrict__ denom, float* __restrict__ pbuf, int E) {
  const int e = blockIdx.x * blockDim.x + threadIdx.x;
  if (e >= E) return;
  const int r = rows[e], c = cols[e];
  const float w = ea[e];
  float si4[4], sj4[4];
  unsigned mu4[4];
  *(float4*)si4 = ((const float4*)s_i)[r];
  *(float4*)sj4 = ((const float4*)s_j)[c];
  *(uint4*)mu4 = ((const uint4*)mseg)[c];
  float p4[4];
  #pragma unroll
  for (int hh = 0; hh < HEADS; ++hh) {
    const float coef = linE[hh] * att[hh * ATT_ROW + 2 * HID];
    float lg = si4[hh] + sj4[hh] + w * coef;
    lg = (lg >= 0.f) ? lg : NEG_SLOPE * lg;
    const float p = expf(lg - funmap(mu4[hh]));
    p4[hh] = p;
    atomicAdd(&denom[(size_t)c * HEADS + hh], p);
  }
  ((float4*)pbuf)[e] = *(float4*)p4;
}

// ---------- edge pass 3: acc[col,c] += sum_h alpha_h * h[row,h,c] ----------
// 4 edges per 256-thread block; 64 lanes per edge (one per channel).
__global__ __launch_bounds__(256) void edge_msg(
    const int* __restrict__ rows, const int* __restrict__ cols,
    const float* __restrict__ pbuf, const float* __restrict__ denom,
    const float* __restrict__ h, float* __restrict__ acc, int E) {
  __shared__ float s_alpha[4][HEADS];
  __shared__ int s_rc[4][2];
  const int t = threadIdx.x;
  const int el = t >> 6;               // edge slot 0..3
  const int c  = t & 63;               // channel
  const int e = blockIdx.x * 4 + el;
  if (e < E && c < HEADS) {
    const int col = cols[e];
    const float p = pbuf[(size_t)e * HEADS + c];
    const float d = denom[(size_t)col * HEADS + c];
    s_alpha[el][c] = p / (d + 1e-16f);
    if (c == 0) { s_rc[el][0] = rows[e]; s_rc[el][1] = col; }
  }
  __syncthreads();
  if (e < E) {
    const int r = s_rc[el][0], col = s_rc[el][1];
    const float* hr = h + (size_t)r * HC + c;
    const float val = s_alpha[el][0] * hr[0]
                    + s_alpha[el][1] * hr[HID]
                    + s_alpha[el][2] * hr[2 * HID]
                    + s_alpha[el][3] * hr[3 * HID];
    atomicAdd(acc + (size_t)col * HID + c, val);
  }
}

// ---------- node epilogue: relu(acc/H + bias + mean_h(h)) ----------
__global__ void node_fin(const float* __restrict__ h, const float* __restrict__ acc,
                         const float* __restrict__ bias, float* __restrict__ nxt, int N) {
  const size_t idx = (size_t)blockIdx.x * blockDim.x + threadIdx.x;
  if (idx >= (size_t)N * HID) return;
  const int n = (int)(idx >> 6);
  const int c = (int)(idx & 63);
  const float* hr = h + (size_t)n * HC + c;
  const float hm = 0.25f * (hr[0] + hr[HID] + hr[2 * HID] + hr[3 * HID]);
  const float v = acc[idx] * 0.25f + bias[c] + hm;
  nxt[idx] = fmaxf(v, 0.f);
}

extern "C" void kernel_launch(void* const* d_in, const int* in_sizes, int n_in,
                              void* d_out, int out_size, void* d_ws, size_t ws_size,
                              hipStream_t stream) {
  const float* x        = (const float*)d_in[0];
  const int*   ei       = (const int*)  d_in[1];
  const float* ea       = (const float*)d_in[2];
  const float* linN0    = (const float*)d_in[3];
  const float* linNr    = (const float*)d_in[4];
  const float* linE_all = (const float*)d_in[5];
  const float* att_all  = (const float*)d_in[6];
  const float* bias_all = (const float*)d_in[7];
  const float* bn_gamma = (const float*)d_in[8];
  const float* bn_beta  = (const float*)d_in[9];
  const float* W        = (const float*)d_in[10];
  const float* bvec     = (const float*)d_in[11];
  float* out = (float*)d_out;

  const int N = in_sizes[0] / DIN;
  const int E = in_sizes[1] / 2;
  const int* rows = ei;
  const int* cols = ei + E;

  // -------- workspace layout (floats) --------
  float* ws = (float*)d_ws;
  float* h       = ws; ws += (size_t)N * HC;       // N*256
  float* s_i     = ws; ws += (size_t)N * HEADS;
  float* s_j     = ws; ws += (size_t)N * HEADS;
  unsigned* mseg = (unsigned*)ws; ws += (size_t)N * HEADS;  // contiguous with denom
  float* denom   = ws; ws += (size_t)N * HEADS;
  float* pbuf    = ws; ws += (size_t)E * HEADS;
  float* acc     = ws; ws += (size_t)N * HID;
  float* embA    = ws; ws += (size_t)N * HID;
  float* embB    = ws; ws += (size_t)N * HID;
  float* bnsum   = ws; ws += 2 * HID;
  float* bnscale = ws; ws += HID;
  float* bnshift = ws; ws += HID;

  const dim3 blk(256);
  const int mtiles = (N + 15) / 16;

  for (int l = 0; l < 3; ++l) {
    const float* cur = (l == 0) ? x : ((l == 1) ? embA : embB);
    float* nxt       = (l == 0) ? embA : ((l == 1) ? embB : embA);
    const int K = (l == 0) ? DIN : HID;
    const float* Bmat = (l == 0) ? linN0 : (linNr + (size_t)(l - 1) * HID * HC);
    // HC=256 -> 4 strips/row-tile -> 2 row-tiles per 8-wave block
    const int gemmBlocks = (mtiles + 1) / 2;

    if (l > 0) {
      fill_u32<<<1, 128, 0, stream>>>((unsigned*)bnsum, 0u, 2 * HID);
      bn_reduce<<<512, HID, 0, stream>>>(cur, bnsum, N);
      bn_final<<<1, HID, 0, stream>>>(bnsum, bn_gamma + (size_t)(l - 1) * HID,
                                      bn_beta + (size_t)(l - 1) * HID,
                                      bnscale, bnshift, N);
      gemm_wmma_f32<true><<<gemmBlocks, blk, 0, stream>>>(
          cur, Bmat, h, N, K, HC, bnscale, bnshift, nullptr);
    } else {
      gemm_wmma_f32<false><<<gemmBlocks, blk, 0, stream>>>(
          cur, Bmat, h, N, K, HC, nullptr, nullptr, nullptr);
    }

    node_s<<<(N + 255) / 256, blk, 0, stream>>>(h, att_all + (size_t)l * ATT_L,
                                                s_i, s_j, N);

    // zero segment-max (0u == mapped -NaN), denom (contiguous), and accumulator
    fill_u32<<<2048, 256, 0, stream>>>(mseg, 0u, (size_t)N * HEADS * 2);
    fill_u32<<<2048, 256, 0, stream>>>((unsigned*)acc, 0u, (size_t)N * HID);

    edge_max<<<(E + 255) / 256, blk, 0, stream>>>(
        rows, cols, ea, s_i, s_j,
        linE_all + (size_t)l * HEADS, att_all + (size_t)l * ATT_L, mseg, E);

    edge_p<<<(E + 255) / 256, blk, 0, stream>>>(
        rows, cols, ea, s_i, s_j,
        linE_all + (size_t)l * HEADS, att_all + (size_t)l * ATT_L,
        mseg, denom, pbuf, E);

    edge_msg<<<(E + 3) / 4, blk, 0, stream>>>(rows, cols, pbuf, denom, h, acc, E);

    node_fin<<<(unsigned)(((size_t)N * HID + 255) / 256), blk, 0, stream>>>(
        h, acc, bias_all + (size_t)l * HID, nxt, N);
  }

  // final projection: out = embA @ W + b   (after l=2, result lives in embA)
  // EMB=128 -> 2 strips/row-tile -> 4 row-tiles per 8-wave block
  gemm_wmma_f32<false><<<(mtiles + 3) / 4, blk, 0, stream>>>(
      embA, W, out, N, HID, EMB, nullptr, nullptr, bvec);
}